// ConvLSTM_11630771438090
// MI455X (gfx1250) — compile-verified
//
#include <hip/hip_runtime.h>
#include <math.h>

// ConvLSTM-scan for MI455X (gfx1250, wave32).
// Phase 1: Xp = X @ Wx^T + bias via V_WMMA_F32_16X16X4_F32 (fp32 tensor path).
// Phase 2: per-batch sequential scan, state double-buffered in LDS,
//          exploiting the identity structure of Wup/Wstay/Wdown.

typedef float v2f __attribute__((ext_vector_type(2)));
typedef float v8f __attribute__((ext_vector_type(8)));

#define T_DIM 128
#define B_DIM 32
#define E_DIM 128
#define H_DIM 128
#define GE_DIM 256          // E + H (gate input width)
#define G_DIM 768           // 3H (f) + H (i) + H (c) + H (o)
#define GI_OFF 384
#define GC_OFF 512
#define GO_OFF 640
#define ROWS (T_DIM * B_DIM)  // 4096
#define SSTR (T_DIM + 1)      // LDS state row stride (bank-conflict pad)

// ---------------------------------------------------------------------------
// Row/bias lookup across the four packed weight matrices (gate index k in [0,768))
// k in [0,384): Wf row k ; [384,512): Wi ; [512,640): Wc ; [640,768): Wo
// ---------------------------------------------------------------------------
__device__ __forceinline__ const float* wrow(int k,
    const float* Wf, const float* Wi, const float* Wc, const float* Wo) {
  if (k < GI_OFF) return Wf + (size_t)k * GE_DIM;
  if (k < GC_OFF) return Wi + (size_t)(k - GI_OFF) * GE_DIM;
  if (k < GO_OFF) return Wc + (size_t)(k - GC_OFF) * GE_DIM;
  return Wo + (size_t)(k - GO_OFF) * GE_DIM;
}
__device__ __forceinline__ float biasAt(int k,
    const float* bf, const float* bi, const float* bc, const float* bo) {
  if (k < GI_OFF) return bf[k];
  if (k < GC_OFF) return bi[k - GI_OFF];
  if (k < GO_OFF) return bc[k - GC_OFF];
  return bo[k - GO_OFF];
}

// ---------------------------------------------------------------------------
// Phase 1: Xp[r][k] = sum_e x[r][e] * W[k][H+e] + bias[k]
//   r = t*B + b over 4096 rows, k over 768 gate outputs.
//   One wave per 16x16 output tile, K swept 4 at a time with f32 WMMA.
//   A (16x4): lane L holds M=L%16, K={0,1} (lanes 0-15) / K={2,3} (lanes 16-31)
//   B (4x16): lane L holds N=L%16, same K split (symmetric layout).
// ---------------------------------------------------------------------------
__global__ void __launch_bounds__(256)
xproj_wmma(const float* __restrict__ x,
           const float* __restrict__ Wf, const float* __restrict__ bf,
           const float* __restrict__ Wi, const float* __restrict__ bi,
           const float* __restrict__ Wc, const float* __restrict__ bc,
           const float* __restrict__ Wo, const float* __restrict__ bo,
           float* __restrict__ Xp) {
  const int wid  = blockIdx.x * (blockDim.x >> 5) + (threadIdx.x >> 5);
  const int lane = threadIdx.x & 31;
  const int rowTile = wid / (G_DIM / 16);   // 48 col tiles
  const int colTile = wid % (G_DIM / 16);
  if (rowTile >= ROWS / 16) return;
  const int r0 = rowTile * 16;
  const int k0 = colTile * 16;
  const int mn = lane & 15;                 // M for A, N for B
  const int kb = (lane < 16) ? 0 : 2;       // K sub-offset per lane half

  const float* xrow = x + (size_t)(r0 + mn) * E_DIM + kb;
  const int kg = k0 + mn;                   // global gate column for this lane
  const float* wr = wrow(kg, Wf, Wi, Wc, Wo) + H_DIM + kb;  // x-part columns
  const float bb = biasAt(kg, bf, bi, bc, bo);

  v8f acc0 = {};
  v8f acc1 = {};
  #pragma unroll
  for (int kk = 0; kk < E_DIM; kk += 8) {
    v2f a0 = *(const v2f*)(xrow + kk);
    v2f b0 = *(const v2f*)(wr + kk);
    v2f a1 = *(const v2f*)(xrow + kk + 4);
    v2f b1 = *(const v2f*)(wr + kk + 4);
    acc0 = __builtin_amdgcn_wmma_f32_16x16x4_f32(false, a0, false, b0,
                                                 (short)0, acc0, false, false);
    acc1 = __builtin_amdgcn_wmma_f32_16x16x4_f32(false, a1, false, b1,
                                                 (short)0, acc1, false, false);
  }
  const int mbase = (lane < 16) ? 0 : 8;    // C/D: VGPR v holds M = v (+8 hi half)
  #pragma unroll
  for (int v = 0; v < 8; ++v) {
    Xp[(size_t)(r0 + mbase + v) * G_DIM + kg] = acc0[v] + acc1[v] + bb;
  }
}

// ---------------------------------------------------------------------------
// Phase 1b: WhT[i][k] = W[k][i] for i in [0,128) — h-part, transposed so the
// scan's matvec reads are coalesced across the 768 gate outputs.
// ---------------------------------------------------------------------------
__global__ void wh_transpose(const float* __restrict__ Wf, const float* __restrict__ Wi,
                             const float* __restrict__ Wc, const float* __restrict__ Wo,
                             float* __restrict__ WhT) {
  int g = blockIdx.x * blockDim.x + threadIdx.x;
  if (g >= H_DIM * G_DIM) return;
  int i = g / G_DIM;
  int k = g - i * G_DIM;
  WhT[g] = wrow(k, Wf, Wi, Wc, Wo)[i];
}

// ---------------------------------------------------------------------------
// Phase 2: sequential scan. One workgroup per batch element (32 WGs), 256
// threads (8 wave32's). State (H x T) double-buffered in LDS. Per step:
//   gates = Xp[t,b] + h @ Wh^T      (coalesced matvec vs WhT)
//   softmax3 / sigmoid / tanh       (128 lanes)
//   state update: new = sl*f0 + (sm+b_stay)*f1 + (sr+b_down)*f2
//   h = state[:,t] * o ; emit to out
// ---------------------------------------------------------------------------
__global__ void __launch_bounds__(256)
scan_kernel(const float* __restrict__ Xp,
            const float* __restrict__ WhT,
            const float* __restrict__ b_stay,
            const float* __restrict__ b_down,
            float* __restrict__ out) {
  extern __shared__ float smem[];
  float* cur   = smem;                        // H * SSTR
  float* nxt   = cur + H_DIM * SSTR;          // H * SSTR
  float* gates = nxt + H_DIM * SSTR;          // 768
  float* fa = gates + G_DIM;                  // 128
  float* fb = fa + H_DIM;
  float* fc = fb + H_DIM;
  float* ov = fc + H_DIM;
  float* hv = ov + H_DIM;
  float* bs = hv + H_DIM;
  float* bd = bs + H_DIM;

  const int tid = threadIdx.x;
  const int b   = blockIdx.x;

  for (int i = tid; i < H_DIM * SSTR; i += blockDim.x) { cur[i] = 0.f; nxt[i] = 0.f; }
  if (tid < H_DIM) { hv[tid] = 0.f; bs[tid] = b_stay[tid]; bd[tid] = b_down[tid]; }
  __syncthreads();

  for (int t = 0; t < T_DIM; ++t) {
    const float* xp = Xp + (size_t)(t * B_DIM + b) * G_DIM;
    if (t + 1 < T_DIM) {  // pre-pull next step's projected inputs (global_prefetch_b8)
      const float* xpn = xp + (size_t)B_DIM * G_DIM;
      __builtin_prefetch(xpn + tid, 0, 0);
      __builtin_prefetch(xpn + tid + 256, 0, 0);
      __builtin_prefetch(xpn + tid + 512, 0, 0);
    }
    // ---- gates: thread owns k = tid, tid+256, tid+512 -----------------
    #pragma unroll
    for (int q = 0; q < 3; ++q) {
      const int k = tid + q * 256;
      float acc = xp[k];
      #pragma unroll 8
      for (int i = 0; i < H_DIM; ++i) acc = fmaf(hv[i], WhT[i * G_DIM + k], acc);
      gates[k] = acc;
    }
    __syncthreads();
    // ---- nonlinearities + write new column t --------------------------
    if (tid < H_DIM) {
      float r0 = gates[tid * 3 + 0], r1 = gates[tid * 3 + 1], r2 = gates[tid * 3 + 2];
      float mx = fmaxf(r0, fmaxf(r1, r2));
      float e0 = __expf(r0 - mx), e1 = __expf(r1 - mx), e2 = __expf(r2 - mx);
      float inv = 1.f / (e0 + e1 + e2);
      fa[tid] = e0 * inv; fb[tid] = e1 * inv; fc[tid] = e2 * inv;
      float iv = 1.f / (1.f + __expf(-gates[GI_OFF + tid]));
      float cv = tanhf(gates[GC_OFF + tid]);
      ov[tid]  = 1.f / (1.f + __expf(-gates[GO_OFF + tid]));
      cur[tid * SSTR + t] = iv * cv;
    }
    __syncthreads();
    // ---- state update (identity conv3 structure) ----------------------
    for (int e = tid; e < H_DIM * T_DIM; e += blockDim.x) {
      const int hh = e >> 7;          // / T_DIM
      const int j  = e & (T_DIM - 1);
      const int row = hh * SSTR;
      const int jm = (j > 0) ? j - 1 : 0;
      int jp = j + 1; if (jp > t) jp = t;     // min(j+1, t)
      const float sl = cur[row + jm];
      const float sm = cur[row + j];
      const float sr = cur[row + jp];
      nxt[row + j] = sl * fa[hh] + (sm + bs[hh]) * fb[hh] + (sr + bd[hh]) * fc[hh];
    }
    __syncthreads();
    // ---- output gate + emit h ----------------------------------------
    if (tid < H_DIM) {
      float hn = nxt[tid * SSTR + t] * ov[tid];
      hv[tid] = hn;
      out[(size_t)(t * B_DIM + b) * H_DIM + tid] = hn;
    }
    __syncthreads();
    float* tmp = cur; cur = nxt; nxt = tmp;
  }
  // final h (second tuple output)
  if (tid < H_DIM) {
    out[(size_t)T_DIM * B_DIM * H_DIM + (size_t)b * H_DIM + tid] = hv[tid];
  }
}

// ---------------------------------------------------------------------------
// Host entry
// Inputs: 0:x 1:Wf 2:bf 3:Wi 4:bi 5:Wc 6:bc 7:Wo 8:bo 9:Wup 10:Wstay
//         11:b_stay 12:Wdown 13:b_down
// ---------------------------------------------------------------------------
extern "C" void kernel_launch(void* const* d_in, const int* in_sizes, int n_in,
                              void* d_out, int out_size, void* d_ws, size_t ws_size,
                              hipStream_t stream) {
  (void)in_sizes; (void)n_in; (void)out_size; (void)ws_size;
  const float* x  = (const float*)d_in[0];
  const float* Wf = (const float*)d_in[1];
  const float* bf = (const float*)d_in[2];
  const float* Wi = (const float*)d_in[3];
  const float* bi = (const float*)d_in[4];
  const float* Wc = (const float*)d_in[5];
  const float* bc = (const float*)d_in[6];
  const float* Wo = (const float*)d_in[7];
  const float* bo = (const float*)d_in[8];
  const float* b_stay = (const float*)d_in[11];
  const float* b_down = (const float*)d_in[13];
  float* out = (float*)d_out;

  float* Xp  = (float*)d_ws;                     // 4096*768 f32 = 12.0 MB
  float* WhT = Xp + (size_t)ROWS * G_DIM;        // 128*768  f32 = 0.38 MB

  // Phase 1: 12288 tiles, 8 waves/block -> 1536 blocks
  xproj_wmma<<<(ROWS / 16) * (G_DIM / 16) / 8, 256, 0, stream>>>(
      x, Wf, bf, Wi, bi, Wc, bc, Wo, bo, Xp);
  wh_transpose<<<(H_DIM * G_DIM + 255) / 256, 256, 0, stream>>>(Wf, Wi, Wc, Wo, WhT);

  // Phase 2: 32 independent per-batch scans; ~136 KB LDS each (<< 320 KB/WGP)
  size_t shmem = (size_t)(2 * H_DIM * SSTR + G_DIM + 7 * H_DIM) * sizeof(float);
  (void)hipFuncSetAttribute((const void*)scan_kernel,
                            hipFuncAttributeMaxDynamicSharedMemorySize, (int)shmem);
  scan_kernel<<<B_DIM, 256, shmem, stream>>>(Xp, WhT, b_stay, b_down, out);
}